// PointNetFeaturePropagation_59717225284068
// MI455X (gfx1250) — compile-verified
//
#include <hip/hip_runtime.h>
#include <hip/hip_bf16.h>
#include <float.h>
#include <stdint.h>

typedef float v2f __attribute__((ext_vector_type(2)));
typedef float v8f __attribute__((ext_vector_type(8)));

#define B_   16
#define N_   4096
#define S_   1024
#define D1_  128
#define D2_  256
#define C0_  384   // Cin of layer 0 (128 + 256)
#define M_   256   // out channels, both layers

#if __has_builtin(__builtin_amdgcn_s_wait_asynccnt)
#define ASYNC_WAIT(n) __builtin_amdgcn_s_wait_asynccnt(n)
#else
#define ASYNC_WAIT(n) asm volatile("s_wait_asynccnt %0" ::"i"(n) : "memory")
#endif

// Async DMA: global (64-bit VGPR addr) -> LDS (per-lane byte offset in VGPR).
// ISA 15.18.3 opcode 98, tracked by ASYNCcnt. LDS offset = low 32 bits of the
// generic pointer (flat aperture truncates to addr[31:0] for LDS).
#define ASYNC_LOAD_B128(gptr, lptr)                                            \
    asm volatile("global_load_async_to_lds_b128 %0, %1, off"                   \
                 ::"v"((uint32_t)(uintptr_t)(lptr)),                           \
                   "v"((const void*)(gptr))                                    \
                 : "memory")

// ---------------------------------------------------------------------------
// Fold conv bias + eval-mode BN into per-channel scale/shift:
//   y_out = relu( y*scale + shift ),  scale = g/sqrt(v+eps),
//   shift = beta + (bias - mean)*scale
// ---------------------------------------------------------------------------
__global__ __launch_bounds__(256) void bnprep_kernel(
    const float* __restrict__ b0, const float* __restrict__ g0,
    const float* __restrict__ be0, const float* __restrict__ m0,
    const float* __restrict__ v0,
    const float* __restrict__ b1, const float* __restrict__ g1,
    const float* __restrict__ be1, const float* __restrict__ m1,
    const float* __restrict__ v1,
    float* __restrict__ scale, float* __restrict__ shift)   // [2][256]
{
    int t = blockIdx.x * 256 + threadIdx.x;
    if (t >= 2 * M_) return;
    int layer = t >> 8, c = t & 255;
    const float* bb = layer ? b1 : b0;
    const float* gg = layer ? g1 : g0;
    const float* be = layer ? be1 : be0;
    const float* mm = layer ? m1 : m0;
    const float* vv = layer ? v1 : v0;
    float sc = gg[c] * rsqrtf(vv[c] + 1e-5f);
    scale[t] = sc;
    shift[t] = be[c] + (bb[c] - mm[c]) * sc;
}

// ---------------------------------------------------------------------------
// 3-NN: for each (b,n) scan all S source points (xyz2 staged in LDS),
// keep 3 smallest squared distances, emit normalized inverse-distance weights.
// ---------------------------------------------------------------------------
__global__ __launch_bounds__(256) void knn3_kernel(
    const float* __restrict__ xyz1, const float* __restrict__ xyz2,
    int* __restrict__ idx3, float* __restrict__ w3)
{
    __shared__ float s2[S_ * 3];
    const int b = blockIdx.y;
    const int tid = threadIdx.x;
    for (int i = tid; i < S_ * 3; i += 256) s2[i] = xyz2[(size_t)b * S_ * 3 + i];
    __syncthreads();

    const int n = blockIdx.x * 256 + tid;
    const float* p = xyz1 + ((size_t)b * N_ + n) * 3;
    const float px = p[0], py = p[1], pz = p[2];

    float d0 = FLT_MAX, d1 = FLT_MAX, d2 = FLT_MAX;
    int i0 = 0, i1 = 0, i2 = 0;
    for (int s = 0; s < S_; ++s) {
        float dx = px - s2[3 * s + 0];
        float dy = py - s2[3 * s + 1];
        float dz = pz - s2[3 * s + 2];
        float d = dx * dx + dy * dy + dz * dz;
        if (d < d2) {
            if (d < d1) {
                if (d < d0) { d2 = d1; i2 = i1; d1 = d0; i1 = i0; d0 = d; i0 = s; }
                else        { d2 = d1; i2 = i1; d1 = d;  i1 = s; }
            } else          { d2 = d;  i2 = s; }
        }
    }
    float r0 = 1.f / (d0 + 1e-8f);
    float r1 = 1.f / (d1 + 1e-8f);
    float r2 = 1.f / (d2 + 1e-8f);
    float inv = 1.f / (r0 + r1 + r2);
    size_t base = ((size_t)b * N_ + n) * 3;
    idx3[base + 0] = i0; idx3[base + 1] = i1; idx3[base + 2] = i2;
    w3[base + 0] = r0 * inv; w3[base + 1] = r1 * inv; w3[base + 2] = r2 * inv;
}

// ---------------------------------------------------------------------------
// Tiled transpose: per batch, src is [C, Ncols]; write dst[n*dstRowStride + c].
// ---------------------------------------------------------------------------
__global__ void transpose_into_kernel(
    const float* __restrict__ src, float* __restrict__ dst,
    int Ncols, int dstRowStride, size_t srcBatchStride, size_t dstBatchStride)
{
    __shared__ float tile[32][33];
    const int b = blockIdx.z;
    const int c0 = blockIdx.y * 32;
    const int n0 = blockIdx.x * 32;
    const int tx = threadIdx.x, ty = threadIdx.y;

    const float* sb = src + (size_t)b * srcBatchStride;
    float* db = dst + (size_t)b * dstBatchStride;

#pragma unroll
    for (int j = 0; j < 4; ++j)
        tile[ty + 8 * j][tx] = sb[(size_t)(c0 + ty + 8 * j) * Ncols + n0 + tx];
    __syncthreads();
#pragma unroll
    for (int j = 0; j < 4; ++j)
        db[(size_t)(n0 + ty + 8 * j) * dstRowStride + c0 + tx] = tile[tx][ty + 8 * j];
}

// ---------------------------------------------------------------------------
// Interpolation: X[b][n][128+c] = sum_k w_k * pts2T[b][idx_k][c]
// pts2T is [B, S, 256]; gathers are coalesced across c.
// ---------------------------------------------------------------------------
__global__ __launch_bounds__(256) void interp_kernel(
    const float* __restrict__ pts2T, const int* __restrict__ idx3,
    const float* __restrict__ w3, float* __restrict__ X)
{
    const int b = blockIdx.y;
    const int n = blockIdx.x;
    const int c = threadIdx.x;
    size_t base = ((size_t)b * N_ + n) * 3;
    int i0 = idx3[base + 0], i1 = idx3[base + 1], i2 = idx3[base + 2];
    float w0 = w3[base + 0], w1 = w3[base + 1], w2 = w3[base + 2];
    const float* p = pts2T + (size_t)b * S_ * D2_;
    float v = w0 * p[(size_t)i0 * D2_ + c]
            + w1 * p[(size_t)i1 * D2_ + c]
            + w2 * p[(size_t)i2 * D2_ + c];
    X[((size_t)b * N_ + n) * C0_ + D1_ + c] = v;
}

// ---------------------------------------------------------------------------
// GEMM + fused BN + ReLU using V_WMMA_F32_16X16X4_F32 (exact fp32).
//   Y[b] = relu( W[256,K] @ Xin[b]^T * scale + shift ),  Xin: [B, N, K]
// Block: 256 threads = 8 waves; 64(M) x 64(N) tile; wave = 16M x 32N
// (two 16x16 subtiles sharing the A fragment).
// K staged 16 wide in LDS via double-buffered GLOBAL_LOAD_ASYNC_TO_LDS_B128
// (ASYNCcnt pipeline); steady-state loop is branch-free, last tile peeled.
// A lane layout (16x4):  lane l -> m=l%16, k = k0 + 2*(l/16) + {0,1}
// B lane layout (4x16):  lane l -> n=l%16, k = k0 + 2*(l/16) + {0,1}
// C/D layout: VGPR r, lane l -> m = r + 8*(l/16), n = l%16
// ---------------------------------------------------------------------------
__global__ __launch_bounds__(256) void gemm_bn_relu_kernel(
    const float* __restrict__ Xin,   // [B, N, K]
    const float* __restrict__ Wt,    // [256, K]
    const float* __restrict__ scale, // [256]
    const float* __restrict__ shift, // [256]
    float* __restrict__ Yout,
    int K, int chmajor)
{
    // row stride 20 floats = 80B: 16B-aligned rows for async B128 LDS writes,
    // 8B-aligned v2f fragment reads, bank step of 4 -> conflict-free b64 reads.
    __shared__ __align__(16) float Ws[2][64][20];
    __shared__ __align__(16) float Xs[2][64][20];

    const int tid  = threadIdx.x;
    const int lane = tid & 31;
    const int wave = tid >> 5;
    const int hi   = lane >> 4;
    const int lo   = lane & 15;

    const int m0 = blockIdx.y * 64;
    const int n0 = blockIdx.x * 64;
    const int b  = blockIdx.z;

    const int msub  = (wave >> 1) * 16;
    const int nsub0 = (wave & 1) * 32;
    const int nsub1 = nsub0 + 16;

    const int lrow = tid >> 2;        // 0..63
    const int lcol = (tid & 3) * 4;   // 0,4,8,12

    const float* wp = Wt + (size_t)(m0 + lrow) * K + lcol;
    const float* xp = Xin + ((size_t)b * N_ + n0 + lrow) * K + lcol;

    // BN scale/shift loads hoisted above the MMA pipeline (overlap w/ loads)
    const int mbase = m0 + msub + hi * 8;       // 8 consecutive channels / lane
    float4 sA = *(const float4*)&scale[mbase];
    float4 sB = *(const float4*)&scale[mbase + 4];
    float4 hA = *(const float4*)&shift[mbase];
    float4 hB = *(const float4*)&shift[mbase + 4];

    v8f c0 = {}; v8f c1 = {};

#define ASYNC_ISSUE(buf, kk0)                                                  \
    do {                                                                       \
        ASYNC_LOAD_B128(wp + (kk0), &Ws[buf][lrow][lcol]);                     \
        ASYNC_LOAD_B128(xp + (kk0), &Xs[buf][lrow][lcol]);                     \
    } while (0)

#define MMA_STEP(buf)                                                          \
    do {                                                                       \
        _Pragma("unroll")                                                      \
        for (int kk = 0; kk < 16; kk += 4) {                                   \
            v2f a   = *(const v2f*)&Ws[buf][msub + lo][kk + 2 * hi];           \
            v2f bb0 = *(const v2f*)&Xs[buf][nsub0 + lo][kk + 2 * hi];          \
            v2f bb1 = *(const v2f*)&Xs[buf][nsub1 + lo][kk + 2 * hi];          \
            c0 = __builtin_amdgcn_wmma_f32_16x16x4_f32(                        \
                     false, a, false, bb0, (short)0, c0, false, false);        \
            c1 = __builtin_amdgcn_wmma_f32_16x16x4_f32(                        \
                     false, a, false, bb1, (short)0, c1, false, false);        \
        }                                                                      \
    } while (0)

    // K >= 32 and K % 16 == 0 for both layers: peel the last tile so the
    // steady-state loop is branch-free.
    int cur = 0;
    ASYNC_ISSUE(0, 0);
    int k0 = 0;
    for (; k0 + 16 < K; k0 += 16) {
        ASYNC_ISSUE(cur ^ 1, k0 + 16);       // prefetch next tile -> other buf
        ASYNC_WAIT(2);                       // in-order: current tile landed
        __syncthreads();
        MMA_STEP(cur);
        __syncthreads();                     // all reads done before buf reuse
        cur ^= 1;
    }
    ASYNC_WAIT(0);                           // last tile
    __syncthreads();
    MMA_STEP(cur);
#undef ASYNC_ISSUE
#undef MMA_STEP

    // ---- fused BN + ReLU epilogue ----
    float sc[8] = {sA.x, sA.y, sA.z, sA.w, sB.x, sB.y, sB.z, sB.w};
    float sh[8] = {hA.x, hA.y, hA.z, hA.w, hB.x, hB.y, hB.z, hB.w};

    float f0[8], f1[8];
#pragma unroll
    for (int r = 0; r < 8; ++r) {
        f0[r] = fmaxf(c0[r] * sc[r] + sh[r], 0.f);
        f1[r] = fmaxf(c1[r] * sc[r] + sh[r], 0.f);
    }

    const int ng0 = n0 + nsub0 + lo;
    const int ng1 = n0 + nsub1 + lo;
    if (!chmajor) {
        // [B, N, 256] : per-lane 2x float4 (32B) stores
        float* y0p = Yout + ((size_t)b * N_ + ng0) * M_ + mbase;
        float* y1p = Yout + ((size_t)b * N_ + ng1) * M_ + mbase;
        *(float4*)(y0p + 0) = make_float4(f0[0], f0[1], f0[2], f0[3]);
        *(float4*)(y0p + 4) = make_float4(f0[4], f0[5], f0[6], f0[7]);
        *(float4*)(y1p + 0) = make_float4(f1[0], f1[1], f1[2], f1[3]);
        *(float4*)(y1p + 4) = make_float4(f1[4], f1[5], f1[6], f1[7]);
    } else {
        // [B, 256, N] : each accumulator row r forms a coalesced 64B segment
#pragma unroll
        for (int r = 0; r < 8; ++r) {
            Yout[((size_t)b * M_ + mbase + r) * N_ + ng0] = f0[r];
            Yout[((size_t)b * M_ + mbase + r) * N_ + ng1] = f1[r];
        }
    }
}

// ---------------------------------------------------------------------------
extern "C" void kernel_launch(void* const* d_in, const int* in_sizes, int n_in,
                              void* d_out, int out_size, void* d_ws, size_t ws_size,
                              hipStream_t stream)
{
    const float* xyz1    = (const float*)d_in[0];
    const float* xyz2    = (const float*)d_in[1];
    const float* points1 = (const float*)d_in[2];
    const float* points2 = (const float*)d_in[3];
    const float* w0  = (const float*)d_in[4];
    const float* b0  = (const float*)d_in[5];
    const float* g0  = (const float*)d_in[6];
    const float* be0 = (const float*)d_in[7];
    const float* m0  = (const float*)d_in[8];
    const float* v0  = (const float*)d_in[9];
    const float* w1  = (const float*)d_in[10];
    const float* b1  = (const float*)d_in[11];
    const float* g1  = (const float*)d_in[12];
    const float* be1 = (const float*)d_in[13];
    const float* m1  = (const float*)d_in[14];
    const float* v1  = (const float*)d_in[15];
    float* out = (float*)d_out;

    float* ws = (float*)d_ws;
    size_t offX     = 0;
    size_t offY0    = offX    + (size_t)B_ * N_ * C0_;  // X   : [B,N,384]
    size_t offP2T   = offY0   + (size_t)B_ * N_ * M_;   // Y0  : [B,N,256]
    size_t offIdx   = offP2T  + (size_t)B_ * S_ * D2_;  // P2T : [B,S,256]
    size_t offW3    = offIdx  + (size_t)B_ * N_ * 3;
    size_t offScale = offW3   + (size_t)B_ * N_ * 3;
    size_t offShift = offScale + 2 * M_;

    float* X      = ws + offX;
    float* Y0     = ws + offY0;
    float* P2T    = ws + offP2T;
    int*   idx3   = (int*)(ws + offIdx);
    float* w3     = ws + offW3;
    float* scaleB = ws + offScale;   // [2][256]
    float* shiftB = ws + offShift;   // [2][256]

    bnprep_kernel<<<2, 256, 0, stream>>>(b0, g0, be0, m0, v0,
                                         b1, g1, be1, m1, v1, scaleB, shiftB);

    knn3_kernel<<<dim3(N_ / 256, B_), 256, 0, stream>>>(xyz1, xyz2, idx3, w3);

    // points1 [B,128,N] -> X[:, :, 0:128]   (dst row stride 384)
    transpose_into_kernel<<<dim3(N_ / 32, D1_ / 32, B_), dim3(32, 8), 0, stream>>>(
        points1, X, N_, C0_, (size_t)D1_ * N_, (size_t)N_ * C0_);

    // points2 [B,256,S] -> P2T [B,S,256]
    transpose_into_kernel<<<dim3(S_ / 32, D2_ / 32, B_), dim3(32, 8), 0, stream>>>(
        points2, P2T, S_, D2_, (size_t)D2_ * S_, (size_t)S_ * D2_);

    interp_kernel<<<dim3(N_, B_), 256, 0, stream>>>(P2T, idx3, w3, X);

    // layer 0: [256,384] @ X^T -> Y0 [B,N,256]
    gemm_bn_relu_kernel<<<dim3(N_ / 64, M_ / 64, B_), 256, 0, stream>>>(
        X, w0, scaleB, shiftB, Y0, C0_, 0);

    // layer 1: [256,256] @ Y0^T -> out [B,256,N]
    gemm_bn_relu_kernel<<<dim3(N_ / 64, M_ / 64, B_), 256, 0, stream>>>(
        Y0, w1, scaleB + M_, shiftB + M_, out, M_, 1);
}